// ImprovedDeformableLocalGraphAttention_59863254172139
// MI455X (gfx1250) — compile-verified
//
#include <hip/hip_runtime.h>
#include <math.h>
#include <float.h>

typedef __attribute__((ext_vector_type(2))) float v2f;
typedef __attribute__((ext_vector_type(8))) float v8f;

#define Bb  4
#define Nn  1024
#define Cc  384
#define KK  10
#define NKp (Nn * KK)      // 10240
#define C2  (2 * Cc)       // 768

// CDNA5 float max-num atomic (IEEE maximumNumber) — fuses the max-over-K.
__device__ __forceinline__ void atomic_max_f32(float* addr, float val) {
    asm volatile("global_atomic_max_num_f32 %0, %1, off"
                 :: "v"(addr), "v"(val) : "memory");
}

// CDNA5 async global->LDS copy (ASYNCcnt-tracked, bypasses VGPRs).
// Generic shared-pointer low 32 bits == wave-relative LDS byte offset.
__device__ __forceinline__ unsigned lds_off(const void* p) {
    return (unsigned)(uintptr_t)p;
}
__device__ __forceinline__ void async_b128(unsigned lds, const void* g) {
    asm volatile("global_load_async_to_lds_b128 %0, %1, off"
                 :: "v"(lds), "v"(g) : "memory");
}
__device__ __forceinline__ void wait_async0() {
    asm volatile("s_wait_asynccnt 0x0" ::: "memory");
}

// ---------------------------------------------------------------------------
// K1: KNN (top-10 by squared distance) + half-bbox scale per (b,n)
// ---------------------------------------------------------------------------
__global__ __launch_bounds__(256) void dlga_knn_kernel(
    const float* __restrict__ q_pos, int* __restrict__ idx,
    float* __restrict__ scale)
{
    __shared__ alignas(16) float P[Nn * 3];
    const int b     = blockIdx.x >> 2;
    const int nbase = (blockIdx.x & 3) * 256;
    // 12 KB position table -> LDS via async b128 (768 chunks, 3 per thread)
    for (int ch = threadIdx.x; ch < (Nn * 3) / 4; ch += 256)
        async_b128(lds_off(&P[ch * 4]), q_pos + b * Nn * 3 + ch * 4);
    wait_async0();
    __syncthreads();

    const int n = nbase + threadIdx.x;
    const float px = P[n * 3 + 0], py = P[n * 3 + 1], pz = P[n * 3 + 2];

    float bd[KK]; int bi[KK];
#pragma unroll
    for (int t = 0; t < KK; ++t) { bd[t] = 1e30f; bi[t] = 0; }

    for (int m = 0; m < Nn; ++m) {
        const float dx = px - P[m * 3 + 0];
        const float dy = py - P[m * 3 + 1];
        const float dz = pz - P[m * 3 + 2];
        const float d  = dx * dx + dy * dy + dz * dz;
        if (d < bd[KK - 1]) {
            bd[KK - 1] = d; bi[KK - 1] = m;
#pragma unroll
            for (int t = KK - 1; t > 0; --t) {
                if (bd[t] < bd[t - 1]) {
                    float td = bd[t]; bd[t] = bd[t - 1]; bd[t - 1] = td;
                    int   ti = bi[t]; bi[t] = bi[t - 1]; bi[t - 1] = ti;
                }
            }
        }
    }

    float mn[3] = { 1e30f, 1e30f, 1e30f }, mx[3] = { -1e30f, -1e30f, -1e30f };
#pragma unroll
    for (int t = 0; t < KK; ++t) {
        idx[(b * Nn + n) * KK + t] = bi[t];
#pragma unroll
        for (int dm = 0; dm < 3; ++dm) {
            const float v = P[bi[t] * 3 + dm];
            mn[dm] = fminf(mn[dm], v);
            mx[dm] = fmaxf(mx[dm], v);
        }
    }
#pragma unroll
    for (int dm = 0; dm < 3; ++dm)
        scale[(b * Nn + n) * 3 + dm] = (mx[dm] - mn[dm]) * 0.5f;
}

// ---------------------------------------------------------------------------
// K2: v_off = q @ Wv + bv   via fp32 WMMA.  M = B*N = 4096 (256 tiles of 16).
// ---------------------------------------------------------------------------
__global__ __launch_bounds__(256) void dlga_voff_kernel(
    const float* __restrict__ q, const float* __restrict__ Wv,
    const float* __restrict__ bv, float* __restrict__ v_off)
{
    __shared__ alignas(16) float At[16][Cc];
    const int row0 = blockIdx.x * 16;
    // 24 KB contiguous A tile -> LDS via async b128 (1536 chunks, 6 per thread)
    for (int ch = threadIdx.x; ch < (16 * Cc) / 4; ch += 256)
        async_b128(lds_off(&At[0][0]) + ch * 16,
                   q + (size_t)row0 * Cc + ch * 4);
    wait_async0();
    __syncthreads();

    const int lane = threadIdx.x & 31;
    const int wv   = threadIdx.x >> 5;
    const int rowA = lane & 15;
    const int kh   = (lane >> 4) * 2;   // 0 or 2
    const int colL = lane & 15;
    const int nb   = wv * 48;

    v8f acc0 = {}, acc1 = {}, acc2 = {};
    for (int k0 = 0; k0 < Cc; k0 += 4) {
        v2f a; a.x = At[rowA][k0 + kh]; a.y = At[rowA][k0 + kh + 1];
        const float* Wp = Wv + (size_t)(k0 + kh) * Cc;
        v2f b0; b0.x = Wp[nb + colL];       b0.y = Wp[Cc + nb + colL];
        acc0 = __builtin_amdgcn_wmma_f32_16x16x4_f32(false, a, false, b0, (short)0, acc0, false, false);
        v2f b1; b1.x = Wp[nb + 16 + colL];  b1.y = Wp[Cc + nb + 16 + colL];
        acc1 = __builtin_amdgcn_wmma_f32_16x16x4_f32(false, a, false, b1, (short)0, acc1, false, false);
        v2f b2; b2.x = Wp[nb + 32 + colL];  b2.y = Wp[Cc + nb + 32 + colL];
        acc2 = __builtin_amdgcn_wmma_f32_16x16x4_f32(false, a, false, b2, (short)0, acc2, false, false);
    }

    const int rofs = (lane >> 4) * 8;
#pragma unroll
    for (int j = 0; j < 8; ++j) {
        const size_t r = (size_t)(row0 + j + rofs) * Cc;
        v_off[r + nb +      colL] = acc0[j] + bv[nb +      colL];
        v_off[r + nb + 16 + colL] = acc1[j] + bv[nb + 16 + colL];
        v_off[r + nb + 32 + colL] = acc2[j] + bv[nb + 32 + colL];
    }
}

// ---------------------------------------------------------------------------
// K3: fused offset MLP.  M = B*N*K = 40960 rows (2560 tiles of 16).
// gather/concat -> WMMA GEMM (K=768) -> LayerNorm -> GELU -> @W2 -> tanh
// -> deformed shift_pos.  h-tile reuses the A-tile LDS (columns 0..383).
// ---------------------------------------------------------------------------
__global__ __launch_bounds__(256) void dlga_offset_mlp_kernel(
    const float* __restrict__ q, const float* __restrict__ q_pos,
    const float* __restrict__ v_off, const int* __restrict__ idx,
    const float* __restrict__ scale,
    const float* __restrict__ W1, const float* __restrict__ b1,
    const float* __restrict__ ln_g, const float* __restrict__ ln_b,
    const float* __restrict__ W2, float* __restrict__ shift_pos)
{
    __shared__ alignas(16) float At[16][C2];  // A tile, reused as h tile (c<384)
    __shared__ float Off[16][3];
    const int row0 = blockIdx.x * 16;

    // Stage A = [gathered v_off | broadcast q], pure copies -> async b128.
    // 16 rows x 192 16B-chunks = 3072 chunks, 12 per thread.
    for (int ch = threadIdx.x; ch < 16 * (C2 / 4); ch += 256) {
        const int r  = ch / (C2 / 4);
        const int cq = (ch % (C2 / 4)) * 4;
        const int gr = row0 + r;
        const int b = gr / NKp, rem = gr % NKp, n = rem / KK, kk = rem % KK;
        const float* src;
        if (cq < Cc) {
            const int s = idx[(b * Nn + n) * KK + kk];
            src = v_off + (size_t)(b * Nn + s) * Cc + cq;
        } else {
            src = q + (size_t)(b * Nn + n) * Cc + (cq - Cc);
        }
        async_b128(lds_off(&At[r][cq]), src);
    }
    wait_async0();
    __syncthreads();

    const int lane = threadIdx.x & 31;
    const int wv   = threadIdx.x >> 5;
    const int rowA = lane & 15;
    const int kh   = (lane >> 4) * 2;
    const int colL = lane & 15;
    const int nb   = wv * 48;

    v8f acc0 = {}, acc1 = {}, acc2 = {};
    for (int k0 = 0; k0 < C2; k0 += 4) {
        v2f a; a.x = At[rowA][k0 + kh]; a.y = At[rowA][k0 + kh + 1];
        const float* Wp = W1 + (size_t)(k0 + kh) * Cc;
        v2f b0; b0.x = Wp[nb + colL];       b0.y = Wp[Cc + nb + colL];
        acc0 = __builtin_amdgcn_wmma_f32_16x16x4_f32(false, a, false, b0, (short)0, acc0, false, false);
        v2f bb1; bb1.x = Wp[nb + 16 + colL]; bb1.y = Wp[Cc + nb + 16 + colL];
        acc1 = __builtin_amdgcn_wmma_f32_16x16x4_f32(false, a, false, bb1, (short)0, acc1, false, false);
        v2f b2; b2.x = Wp[nb + 32 + colL];  b2.y = Wp[Cc + nb + 32 + colL];
        acc2 = __builtin_amdgcn_wmma_f32_16x16x4_f32(false, a, false, b2, (short)0, acc2, false, false);
    }
    __syncthreads();   // everyone done reading A before aliasing it with h

    const int rofs = (lane >> 4) * 8;
#pragma unroll
    for (int j = 0; j < 8; ++j) {
        At[j + rofs][nb +      colL] = acc0[j] + b1[nb +      colL];
        At[j + rofs][nb + 16 + colL] = acc1[j] + b1[nb + 16 + colL];
        At[j + rofs][nb + 32 + colL] = acc2[j] + b1[nb + 32 + colL];
    }
    __syncthreads();

    // LayerNorm + exact GELU, one row per thread (threads 0..15)
    if (threadIdx.x < 16) {
        const int r = threadIdx.x;
        float s = 0.f, s2 = 0.f;
        for (int c = 0; c < Cc; ++c) { const float x = At[r][c]; s += x; s2 += x * x; }
        const float mu  = s * (1.0f / Cc);
        const float var = s2 * (1.0f / Cc) - mu * mu;
        const float inv = rsqrtf(var + 1e-5f);
        for (int c = 0; c < Cc; ++c) {
            float x = (At[r][c] - mu) * inv * ln_g[c] + ln_b[c];
            x = 0.5f * x * (1.0f + erff(x * 0.70710678118654752f));  // exact GELU
            At[r][c] = x;
        }
    }
    __syncthreads();

    // offset = tanh(h @ W2): threads 0..47 each do one (row, dim)
    if (threadIdx.x < 48) {
        const int r = threadIdx.x / 3, dm = threadIdx.x % 3;
        float s = 0.f;
        for (int c = 0; c < Cc; ++c) s += At[r][c] * W2[c * 3 + dm];
        Off[r][dm] = tanhf(s);
    }
    __syncthreads();

    // shift_pos = local_v_pos + offset * scale
    if (threadIdx.x < 16) {
        const int r = threadIdx.x;
        const int gr = row0 + r;
        const int b = gr / NKp, rem = gr % NKp, n = rem / KK, kk = rem % KK;
        const int src = idx[(b * Nn + n) * KK + kk];
#pragma unroll
        for (int dm = 0; dm < 3; ++dm) {
            const float p  = q_pos[(b * Nn + src) * 3 + dm];
            const float sc = scale[(b * Nn + n) * 3 + dm];
            shift_pos[(size_t)gr * 3 + dm] = p + Off[r][dm] * sc;
        }
    }
}

// ---------------------------------------------------------------------------
// K4: three_nn + inverse-squared-distance weights
// ---------------------------------------------------------------------------
__global__ __launch_bounds__(256) void dlga_threenn_kernel(
    const float* __restrict__ q_pos, const float* __restrict__ shift_pos,
    float* __restrict__ w3, int* __restrict__ idx3)
{
    __shared__ alignas(16) float P[Nn * 3];
    const int b     = blockIdx.x / 40;
    const int pbase = (blockIdx.x % 40) * 256;
    for (int ch = threadIdx.x; ch < (Nn * 3) / 4; ch += 256)
        async_b128(lds_off(&P[ch * 4]), q_pos + b * Nn * 3 + ch * 4);
    wait_async0();
    __syncthreads();

    const int p  = pbase + threadIdx.x;
    const size_t gp = (size_t)(b * NKp + p);
    const float px = shift_pos[gp * 3 + 0];
    const float py = shift_pos[gp * 3 + 1];
    const float pz = shift_pos[gp * 3 + 2];

    float bd[3] = { 1e30f, 1e30f, 1e30f }; int bi[3] = { 0, 0, 0 };
    for (int m = 0; m < Nn; ++m) {
        const float dx = px - P[m * 3 + 0];
        const float dy = py - P[m * 3 + 1];
        const float dz = pz - P[m * 3 + 2];
        const float d  = dx * dx + dy * dy + dz * dz;
        if (d < bd[2]) {
            bd[2] = d; bi[2] = m;
            if (bd[2] < bd[1]) { float t = bd[1]; bd[1] = bd[2]; bd[2] = t;
                                 int ti = bi[1]; bi[1] = bi[2]; bi[2] = ti; }
            if (bd[1] < bd[0]) { float t = bd[0]; bd[0] = bd[1]; bd[1] = t;
                                 int ti = bi[0]; bi[0] = bi[1]; bi[1] = ti; }
        }
    }
    float w0 = 1.0f / (bd[0] + 1e-8f);
    float w1 = 1.0f / (bd[1] + 1e-8f);
    float w2 = 1.0f / (bd[2] + 1e-8f);
    const float inv = 1.0f / (w0 + w1 + w2);
    w3[gp * 3 + 0] = w0 * inv;  idx3[gp * 3 + 0] = bi[0];
    w3[gp * 3 + 1] = w1 * inv;  idx3[gp * 3 + 1] = bi[1];
    w3[gp * 3 + 2] = w2 * inv;  idx3[gp * 3 + 2] = bi[2];
}

// ---------------------------------------------------------------------------
// K5: output init for the fused atomic max-over-K reduction
// ---------------------------------------------------------------------------
__global__ __launch_bounds__(256) void dlga_init_out_kernel(float* __restrict__ out)
{
    const size_t i = (size_t)blockIdx.x * 256 + threadIdx.x;
    if (i < (size_t)Bb * Nn * Cc) out[i] = -FLT_MAX;
}

// ---------------------------------------------------------------------------
// K6: fused interp + graph conv GEMM + LeakyReLU + atomic max over K
// ---------------------------------------------------------------------------
__global__ __launch_bounds__(256) void dlga_graphconv_kernel(
    const float* __restrict__ q, const float* __restrict__ w3,
    const int* __restrict__ idx3, const float* __restrict__ Wk,
    const float* __restrict__ bk, float* __restrict__ out)
{
    __shared__ alignas(16) float Ft[16][C2];
    const int row0 = blockIdx.x * 16;

    // q-broadcast half (c >= Cc) is a pure copy -> async b128 (1536 chunks)
    for (int ch = threadIdx.x; ch < 16 * (Cc / 4); ch += 256) {
        const int r  = ch / (Cc / 4);
        const int cq = (ch % (Cc / 4)) * 4;
        const int gr = row0 + r;
        const int b = gr / NKp, rem = gr % NKp, n = rem / KK;
        async_b128(lds_off(&Ft[r][Cc + cq]), q + (size_t)(b * Nn + n) * Cc + cq);
    }
    // interp half (c < Cc) needs VALU math -> regular path
    for (int e = threadIdx.x; e < 16 * Cc; e += 256) {
        const int r = e / Cc, c = e % Cc;
        const int gr = row0 + r;
        const int b = gr / NKp, rem = gr % NKp, n = rem / KK;
        float s = 0.f;
#pragma unroll
        for (int t = 0; t < 3; ++t) {
            const int   i3 = idx3[(size_t)gr * 3 + t];
            const float wt = w3[(size_t)gr * 3 + t];
            s += wt * q[(size_t)(b * Nn + i3) * Cc + c];
        }
        Ft[r][c] = s - q[(size_t)(b * Nn + n) * Cc + c];
    }
    wait_async0();
    __syncthreads();

    const int lane = threadIdx.x & 31;
    const int wv   = threadIdx.x >> 5;
    const int rowA = lane & 15;
    const int kh   = (lane >> 4) * 2;
    const int colL = lane & 15;
    const int nb   = wv * 48;

    v8f acc0 = {}, acc1 = {}, acc2 = {};
    for (int k0 = 0; k0 < C2; k0 += 4) {
        v2f a; a.x = Ft[rowA][k0 + kh]; a.y = Ft[rowA][k0 + kh + 1];
        const float* Wp = Wk + (size_t)(k0 + kh) * Cc;
        v2f b0; b0.x = Wp[nb + colL];       b0.y = Wp[Cc + nb + colL];
        acc0 = __builtin_amdgcn_wmma_f32_16x16x4_f32(false, a, false, b0, (short)0, acc0, false, false);
        v2f b1; b1.x = Wp[nb + 16 + colL];  b1.y = Wp[Cc + nb + 16 + colL];
        acc1 = __builtin_amdgcn_wmma_f32_16x16x4_f32(false, a, false, b1, (short)0, acc1, false, false);
        v2f b2; b2.x = Wp[nb + 32 + colL];  b2.y = Wp[Cc + nb + 32 + colL];
        acc2 = __builtin_amdgcn_wmma_f32_16x16x4_f32(false, a, false, b2, (short)0, acc2, false, false);
    }

    const int rofs = (lane >> 4) * 8;
#pragma unroll
    for (int j = 0; j < 8; ++j) {
        const int gr = row0 + j + rofs;
        const int b = gr / NKp, rem = gr % NKp, n = rem / KK;
        float* orow = out + (size_t)(b * Nn + n) * Cc;
#pragma unroll
        for (int t = 0; t < 3; ++t) {
            const int col = nb + t * 16 + colL;
            float v = (t == 0 ? acc0[j] : (t == 1 ? acc1[j] : acc2[j])) + bk[col];
            v = (v > 0.f) ? v : 0.2f * v;               // LeakyReLU(0.2)
            atomic_max_f32(orow + col, v);              // fused max over K
        }
    }
}

// ---------------------------------------------------------------------------
extern "C" void kernel_launch(void* const* d_in, const int* in_sizes, int n_in,
                              void* d_out, int out_size, void* d_ws, size_t ws_size,
                              hipStream_t stream) {
    const float* q     = (const float*)d_in[0];
    const float* q_pos = (const float*)d_in[1];
    const float* Wv    = (const float*)d_in[2];
    const float* bv    = (const float*)d_in[3];
    const float* W1    = (const float*)d_in[4];
    const float* b1    = (const float*)d_in[5];
    const float* ln_g  = (const float*)d_in[6];
    const float* ln_b  = (const float*)d_in[7];
    const float* W2    = (const float*)d_in[8];
    const float* Wk    = (const float*)d_in[9];
    const float* bk    = (const float*)d_in[10];
    float* out = (float*)d_out;

    char* w = (char*)d_ws;
    int*   idx   = (int*)w;    w += (size_t)Bb * Nn * KK * sizeof(int);
    float* scale = (float*)w;  w += (size_t)Bb * Nn * 3 * sizeof(float);
    float* voff  = (float*)w;  w += (size_t)Bb * Nn * Cc * sizeof(float);
    float* spos  = (float*)w;  w += (size_t)Bb * NKp * 3 * sizeof(float);
    float* w3    = (float*)w;  w += (size_t)Bb * NKp * 3 * sizeof(float);
    int*   idx3  = (int*)w;

    dlga_init_out_kernel<<<(Bb * Nn * Cc + 255) / 256, 256, 0, stream>>>(out);
    dlga_knn_kernel<<<Bb * 4, 256, 0, stream>>>(q_pos, idx, scale);
    dlga_voff_kernel<<<(Bb * Nn) / 16, 256, 0, stream>>>(q, Wv, bv, voff);
    dlga_offset_mlp_kernel<<<(Bb * NKp) / 16, 256, 0, stream>>>(
        q, q_pos, voff, idx, scale, W1, b1, ln_g, ln_b, W2, spos);
    dlga_threenn_kernel<<<Bb * 40, 256, 0, stream>>>(q_pos, spos, w3, idx3);
    dlga_graphconv_kernel<<<(Bb * NKp) / 16, 256, 0, stream>>>(
        q, w3, idx3, Wk, bk, out);
}